// Net_20847771255349
// MI455X (gfx1250) — compile-verified
//
#include <hip/hip_runtime.h>
#include <hip/hip_bf16.h>

// ---------------------------------------------------------------------------
// 2-layer GAT (PyG GATConv semantics) for MI455X / gfx1250 (wave32, WMMA).
// GEMMs use V_WMMA_F32_16X16X4_F32 (exact fp32, matches reference numerics).
// Edge phase: segment softmax (max-subtraction omitted: mathematically
// identical, logits are O(1)) + atomic scatter-add aggregation in L2.
// ---------------------------------------------------------------------------

typedef __attribute__((ext_vector_type(8))) float v8f;
typedef __attribute__((ext_vector_type(2))) float v2f;

#define NEG_SLOPE 0.2f

// One wave (32 lanes) computes a 16x16 tile of C = A(MxK) * B(KxN).
// K must be a multiple of 4; M, Ncols multiples of 16 (true here: 100000, 64/32).
// Fragment layouts per CDNA5 ISA 7.12.2:
//   A 16x4  : lane m = lane&15 holds K = koff, koff+1  (koff = 2*(lane>=16))
//   B 4x16  : lane n = lane&15 holds rows K = koff, koff+1
//   C 16x16 : reg r, low lanes -> M=r, high lanes -> M=r+8, col = lane&15
__global__ void gemm16_wmma_f32(const float* __restrict__ A,
                                const float* __restrict__ B,
                                float* __restrict__ C,
                                int M, int K, int Ncols) {
  const int lane = threadIdx.x & 31;
  const int row0 = blockIdx.x * 16;
  const int col0 = blockIdx.y * 16;
  const int m    = lane & 15;
  const int koff = (lane >> 4) << 1;       // 0 or 2

  int arow = row0 + m;
  if (arow >= M) arow = M - 1;             // safety clamp (M divisible by 16 here)

  v8f acc = {};
  for (int k0 = 0; k0 < K; k0 += 4) {
    v2f a, b;
    const int ka = k0 + koff;
    a.x = A[arow * K + ka];
    a.y = A[arow * K + ka + 1];
    b.x = B[(ka    ) * Ncols + col0 + m];
    b.y = B[(ka + 1) * Ncols + col0 + m];
    acc = __builtin_amdgcn_wmma_f32_16x16x4_f32(
        /*neg_a=*/false, a, /*neg_b=*/false, b,
        /*c_mod=*/(short)0, acc, /*reuse_a=*/false, /*reuse_b=*/false);
  }

  const int mbase = (lane >> 4) << 3;      // 0 or 8
#pragma unroll
  for (int r = 0; r < 8; ++r) {
    const int row = row0 + mbase + r;
    if (row < M) C[row * Ncols + col0 + m] = acc[r];
  }
}

// Per-node attention logits: als[n,h] = sum_c h[n,h,c]*a_src[h,c]; same for ald.
template <int HH, int CC>
__global__ void att_logits(const float* __restrict__ h,
                           const float* __restrict__ a_src,
                           const float* __restrict__ a_dst,
                           float* __restrict__ als, float* __restrict__ ald,
                           int Nn) {
  const int t = blockIdx.x * blockDim.x + threadIdx.x;
  if (t >= Nn * HH) return;
  const int hh = t % HH;
  const float* hp = h + t * CC;            // t = n*HH + hh; rows are [HH*CC]
  const float* as = a_src + hh * CC;
  const float* ad = a_dst + hh * CC;
  float s = 0.f, d = 0.f;
#pragma unroll
  for (int c = 0; c < CC; ++c) {
    const float v = hp[c];
    s += v * as[c];
    d += v * ad[c];
  }
  als[t] = s;
  ald[t] = d;
}

template <int HH>
__device__ __forceinline__ float edge_weight(const float* __restrict__ als,
                                             const float* __restrict__ ald,
                                             int s, int d, int hh) {
  float e = als[s * HH + hh] + ald[d * HH + hh];
  e = (e > 0.f) ? e : NEG_SLOPE * e;       // leaky_relu(0.2)
  return __expf(e);                        // max-subtraction omitted (invariant)
}

// Pass 1: denom[dst,h] += exp(leaky_relu(logit)). Self-loops appended.
template <int HH>
__global__ void edge_denom(const int* __restrict__ src, const int* __restrict__ dst,
                           const float* __restrict__ als, const float* __restrict__ ald,
                           float* __restrict__ denom, int Eedges, int Nn) {
  const int t = blockIdx.x * blockDim.x + threadIdx.x;
  const int Etot = Eedges + Nn;
  if (t >= Etot * HH) return;
  const int e = t / HH, hh = t % HH;
  int s, d;
  if (e < Eedges) { s = src[e]; d = dst[e]; } else { s = d = e - Eedges; }
  const float w = edge_weight<HH>(als, ald, s, d, hh);
  atomicAdd(&denom[d * HH + hh], w);
}

// Pass 2: out[dst, h, :] += alpha * h[src, h, :]   (alpha = w / denom[dst,h])
template <int HH, int CC>
__global__ void edge_aggr(const int* __restrict__ src, const int* __restrict__ dst,
                          const float* __restrict__ als, const float* __restrict__ ald,
                          const float* __restrict__ denom,
                          const float* __restrict__ h, float* __restrict__ out,
                          int Eedges, int Nn) {
  const int t = blockIdx.x * blockDim.x + threadIdx.x;
  const int Etot = Eedges + Nn;
  if (t >= Etot * HH) return;
  const int e = t / HH, hh = t % HH;
  int s, d;
  if (e < Eedges) { s = src[e]; d = dst[e]; } else { s = d = e - Eedges; }
  const float w = edge_weight<HH>(als, ald, s, d, hh);
  const float alpha = w / denom[d * HH + hh];
  const float* hp = h + (s * HH + hh) * CC;
  float* op = out + (d * HH + hh) * CC;
#pragma unroll
  for (int c = 0; c < CC; ++c) atomicAdd(&op[c], alpha * hp[c]);
}

__global__ void bias_add(float* __restrict__ out, const float* __restrict__ b,
                         int total, int F) {
  const int t = blockIdx.x * blockDim.x + threadIdx.x;
  if (t >= total) return;
  out[t] += b[t % F];
}

static inline int cdiv(long long a, int b) { return (int)((a + b - 1) / b); }

extern "C" void kernel_launch(void* const* d_in, const int* in_sizes, int n_in,
                              void* d_out, int out_size, void* d_ws, size_t ws_size,
                              hipStream_t stream) {
  const float* x      = (const float*)d_in[0];   // [N,128]
  const int*   ei     = (const int*)  d_in[1];   // [2,E] (src row then dst row)
  const float* W1     = (const float*)d_in[2];   // [128,64]
  const float* a_src1 = (const float*)d_in[3];   // [8,8]
  const float* a_dst1 = (const float*)d_in[4];
  const float* b1     = (const float*)d_in[5];   // [64]
  const float* W2     = (const float*)d_in[6];   // [64,32]
  const float* a_src2 = (const float*)d_in[7];   // [1,32]
  const float* a_dst2 = (const float*)d_in[8];
  const float* b2     = (const float*)d_in[9];   // [32]

  const int N = in_sizes[0] / 128;
  const int E = in_sizes[1] / 2;
  const int* src = ei;
  const int* dst = ei + E;

  const int IN = 128, H1 = 8, C1 = 8, F1 = 64, OUT = 32;

  // Workspace layout (floats)
  float* ws   = (float*)d_ws;
  float* h1     = ws;                 size_t o = (size_t)N * F1;   // [N,64]
  float* al1s   = ws + o;             o += (size_t)N * H1;         // [N,8]
  float* al1d   = ws + o;             o += (size_t)N * H1;
  float* denom1 = ws + o;             o += (size_t)N * H1;
  float* out1   = ws + o;             o += (size_t)N * F1;         // layer-1 output / layer-2 input
  float* h2     = ws + o;             o += (size_t)N * OUT;        // [N,32]
  float* al2s   = ws + o;             o += (size_t)N;
  float* al2d   = ws + o;             o += (size_t)N;
  float* denom2 = ws + o;             o += (size_t)N;
  (void)ws_size;

  const int Etot = E + N;
  float* fout = (float*)d_out;

  // ---------------- Layer 1 ----------------
  {
    dim3 grid(cdiv(N, 16), F1 / 16);
    gemm16_wmma_f32<<<grid, 32, 0, stream>>>(x, W1, h1, N, IN, F1);
  }
  att_logits<H1, C1><<<cdiv((long long)N * H1, 256), 256, 0, stream>>>(
      h1, a_src1, a_dst1, al1s, al1d, N);
  hipMemsetAsync(denom1, 0, (size_t)N * H1 * sizeof(float), stream);
  hipMemsetAsync(out1,   0, (size_t)N * F1 * sizeof(float), stream);
  edge_denom<H1><<<cdiv((long long)Etot * H1, 256), 256, 0, stream>>>(
      src, dst, al1s, al1d, denom1, E, N);
  edge_aggr<H1, C1><<<cdiv((long long)Etot * H1, 256), 256, 0, stream>>>(
      src, dst, al1s, al1d, denom1, h1, out1, E, N);
  bias_add<<<cdiv((long long)N * F1, 256), 256, 0, stream>>>(out1, b1, N * F1, F1);

  // ---------------- Layer 2 ----------------
  {
    dim3 grid(cdiv(N, 16), OUT / 16);
    gemm16_wmma_f32<<<grid, 32, 0, stream>>>(out1, W2, h2, N, F1, OUT);
  }
  att_logits<1, OUT><<<cdiv(N, 256), 256, 0, stream>>>(
      h2, a_src2, a_dst2, al2s, al2d, N);
  hipMemsetAsync(denom2, 0, (size_t)N * sizeof(float), stream);
  hipMemsetAsync(fout,   0, (size_t)N * OUT * sizeof(float), stream);
  edge_denom<1><<<cdiv(Etot, 256), 256, 0, stream>>>(
      src, dst, al2s, al2d, denom2, E, N);
  edge_aggr<1, OUT><<<cdiv(Etot, 256), 256, 0, stream>>>(
      src, dst, al2s, al2d, denom2, h2, fout, E, N);
  bias_add<<<cdiv((long long)N * OUT, 256), 256, 0, stream>>>(fout, b2, N * OUT, OUT);
}